// GNN_10496900071979
// MI455X (gfx1250) — compile-verified
//
#include <hip/hip_runtime.h>
#include <math.h>
#include <stdint.h>

#define F 128
#define BN_EPS 1e-5f

typedef __attribute__((ext_vector_type(2))) float v2f;
typedef __attribute__((ext_vector_type(8))) float v8f;

// ---------------------------------------------------------------------------
// CDNA5 async global->LDS copy (ASYNCcnt path, per-lane addressing)
// ---------------------------------------------------------------------------
__device__ __forceinline__ void async_b128_to_lds(uint32_t lds_off, const float* gaddr) {
  asm volatile("global_load_async_to_lds_b128 %0, %1, off"
               :: "v"(lds_off), "v"(gaddr) : "memory");
}
__device__ __forceinline__ void wait_asynccnt0() {
  asm volatile("s_wait_asynccnt 0x0" ::: "memory");
}
// generic LDS pointer -> raw LDS byte offset (ISA 10.2: low 32 bits of the address)
__device__ __forceinline__ uint32_t lds_off_of(const void* p) {
  return (uint32_t)(uintptr_t)p;
}

// ---------------------------------------------------------------------------
// init: agg_n = agg_d = x  (implements h = agg + x with eps=0), zero stats
// ---------------------------------------------------------------------------
__global__ void init_kernel(const float* __restrict__ x, float* __restrict__ a0,
                            float* __restrict__ a1, float* __restrict__ stats,
                            long long total) {
  long long i0 = (long long)blockIdx.x * blockDim.x + threadIdx.x;
  if (i0 < 1024) stats[i0] = 0.f;
  long long stride = (long long)gridDim.x * blockDim.x;
  for (long long i = i0; i < total; i += stride) {
    float v = x[i];
    a0[i] = v;
    a1[i] = v;
  }
}

// ---------------------------------------------------------------------------
// edge scatter: one thread per (edge, 4-float chunk). x rows fit in the
// 192MB L2, so both gathers and the f32 atomics are L2-resident.
// ---------------------------------------------------------------------------
__global__ void scatter_kernel(const float* __restrict__ x, const int* __restrict__ ei,
                               const int* __restrict__ et, float* __restrict__ aggn,
                               float* __restrict__ aggd, int E) {
  long long t0 = (long long)blockIdx.x * blockDim.x + threadIdx.x;
  long long total = (long long)E * 32;
  long long stride = (long long)gridDim.x * blockDim.x;
  for (long long t = t0; t < total; t += stride) {
    int e = (int)(t >> 5);
    int q = (int)(t & 31);
    int src = ei[e];
    int dst = ei[E + e];
    const float4 v = *(const float4*)(x + (size_t)src * F + q * 4);
    float* outp = ((et[e] == 0) ? aggn : aggd) + (size_t)dst * F + q * 4;
    atomicAdd(outp + 0, v.x);
    atomicAdd(outp + 1, v.y);
    atomicAdd(outp + 2, v.z);
    atomicAdd(outp + 3, v.w);
  }
}

// ---------------------------------------------------------------------------
// fp32 WMMA tile accumulation, A sourced from LDS (ds_load), B (=W^T) from
// global. A-frag (ISA 7.12.2): lanes 0-15 -> K={0,1}, lanes 16-31 -> K={2,3}.
// B-frag: VGPR v, lanes 0-15 -> K=v, lanes 16-31 -> K=v+2.
// ---------------------------------------------------------------------------
__device__ __forceinline__ void gemm_tile_accum_lds(const __shared__ float* sA,
                                                    int abase,
                                                    const float* __restrict__ wp,
                                                    v8f& acc) {
#pragma unroll 8
  for (int k = 0; k < F; k += 4) {
    v2f a;
    a.x = sA[abase + k];
    a.y = sA[abase + k + 1];
    v2f b = *(const v2f*)(wp + k);
    acc = __builtin_amdgcn_wmma_f32_16x16x4_f32(false, a, false, b, (short)0, acc,
                                                false, false);
  }
}

// C[M,F] = A[M,F] @ W[F,F]^T + bias ; block stages a 16x128 A tile in LDS via
// async copies, wave w computes output cols [16w, 16w+16).
__global__ void gemm128_kernel(const float* __restrict__ A, const float* __restrict__ W,
                               const float* __restrict__ bias, float* __restrict__ C,
                               int M) {
  __shared__ float smA[16 * F];   // 8 KB
  int tid = threadIdx.x;
  int wave = tid >> 5;
  int lane = tid & 31;
  int lr = lane & 15;
  int hi = lane >> 4;
  int row0 = blockIdx.x * 16;
  int n0 = wave * 16;
  if (row0 >= M) return;

  // stage A tile: 2048 floats = 256 threads x 2 async b128 each
  const float* gA = A + (size_t)row0 * F;
  uint32_t lbase = lds_off_of(&smA[0]);
  async_b128_to_lds(lbase + tid * 16u, gA + tid * 4);
  async_b128_to_lds(lbase + 4096u + tid * 16u, gA + 1024 + tid * 4);
  wait_asynccnt0();
  __syncthreads();

  v8f acc = {};
  int abase = lr * F + 2 * hi;
  const float* wp = W + (size_t)(n0 + lr) * F + 2 * hi;
  gemm_tile_accum_lds(smA, abase, wp, acc);

  int col = n0 + lr;
  float bv = bias[col];
#pragma unroll
  for (int v = 0; v < 8; ++v)
    C[(size_t)(row0 + v + 8 * hi) * F + col] = acc[v] + bv;
}

// logits[M,F] = concat(A0,A1,A2)[M,3F] @ Wg[F,3F]^T + bias  (K=384, no cat buf)
__global__ void gemm_gate_kernel(const float* __restrict__ A0, const float* __restrict__ A1,
                                 const float* __restrict__ A2, const float* __restrict__ W,
                                 const float* __restrict__ bias, float* __restrict__ C,
                                 int M) {
  __shared__ float smA[3 * 16 * F];   // 24 KB
  int tid = threadIdx.x;
  int wave = tid >> 5;
  int lane = tid & 31;
  int lr = lane & 15;
  int hi = lane >> 4;
  int row0 = blockIdx.x * 16;
  int n0 = wave * 16;
  if (row0 >= M) return;

  size_t gbase = (size_t)row0 * F;
  uint32_t lbase = lds_off_of(&smA[0]);
  const float* seg[3] = {A0 + gbase, A1 + gbase, A2 + gbase};
#pragma unroll
  for (int s = 0; s < 3; ++s) {
    async_b128_to_lds(lbase + (uint32_t)s * 8192u + tid * 16u, seg[s] + tid * 4);
    async_b128_to_lds(lbase + (uint32_t)s * 8192u + 4096u + tid * 16u,
                      seg[s] + 1024 + tid * 4);
  }
  wait_asynccnt0();
  __syncthreads();

  v8f acc = {};
  int abase = lr * F + 2 * hi;
  const float* wbase = W + (size_t)(n0 + lr) * (3 * F) + 2 * hi;
  gemm_tile_accum_lds(smA, abase, wbase, acc);
  gemm_tile_accum_lds(smA, 2048 + abase, wbase + F, acc);
  gemm_tile_accum_lds(smA, 4096 + abase, wbase + 2 * F, acc);

  int col = n0 + lr;
  float bv = bias[col];
#pragma unroll
  for (int v = 0; v < 8; ++v)
    C[(size_t)(row0 + v + 8 * hi) * F + col] = acc[v] + bv;
}

// ---------------------------------------------------------------------------
// column sums / sums-of-squares (for BN batch statistics)
// ---------------------------------------------------------------------------
__global__ void col_stats_kernel(const float* __restrict__ h, float* __restrict__ stats,
                                 int M) {
  int c = threadIdx.x;                 // 128 threads = 128 columns
  int r0 = blockIdx.x * 64;
  int rend = r0 + 64 < M ? r0 + 64 : M;
  float s = 0.f, ss = 0.f;
  for (int r = r0; r < rend; ++r) {
    float v = h[(size_t)r * F + c];
    s += v;
    ss += v * v;
  }
  atomicAdd(&stats[c], s);
  atomicAdd(&stats[F + c], ss);
}

// BN (training-mode, biased var) + ReLU, in place
__global__ void bn_relu_kernel(float* __restrict__ h, const float* __restrict__ stats,
                               const float* __restrict__ gamma,
                               const float* __restrict__ beta, long long total,
                               float invM) {
  long long i0 = (long long)blockIdx.x * blockDim.x + threadIdx.x;
  long long stride = (long long)gridDim.x * blockDim.x;
  for (long long i = i0; i < total; i += stride) {
    int c = (int)(i & (F - 1));
    float mu = stats[c] * invM;
    float var = stats[F + c] * invM - mu * mu;
    float v = h[i];
    v = gamma[c] * (v - mu) * rsqrtf(var + BN_EPS) + beta[c];
    h[i] = v > 0.f ? v : 0.f;
  }
}

// ---------------------------------------------------------------------------
// per-row softmax -> inclusive cumsum -> gated combine with flipped x_new_d
// out = (xx + xn) + flip(xd) * gat
// ---------------------------------------------------------------------------
__global__ void finalize_kernel(const float* __restrict__ logits,
                                const float* __restrict__ xx, const float* __restrict__ xn,
                                const float* __restrict__ xd, float* __restrict__ out) {
  __shared__ float sm[F];
  int c = threadIdx.x;
  size_t base = (size_t)blockIdx.x * F;

  float l = logits[base + c];
  sm[c] = l;
  __syncthreads();
  for (int off = 64; off > 0; off >>= 1) {
    if (c < off) sm[c] = fmaxf(sm[c], sm[c + off]);
    __syncthreads();
  }
  float mx = sm[0];
  __syncthreads();

  float p = __expf(l - mx);
  sm[c] = p;
  __syncthreads();
  for (int off = 64; off > 0; off >>= 1) {
    if (c < off) sm[c] += sm[c + off];
    __syncthreads();
  }
  float inv = 1.f / sm[0];
  __syncthreads();
  p *= inv;

  // Hillis-Steele inclusive scan over 128 lanes
  sm[c] = p;
  __syncthreads();
  for (int off = 1; off < F; off <<= 1) {
    float t = (c >= off) ? sm[c - off] : 0.f;
    __syncthreads();
    sm[c] += t;
    __syncthreads();
  }
  float gat = sm[c];

  out[base + c] = xx[base + c] + xn[base + c] + xd[base + (F - 1 - c)] * gat;
}

// ---------------------------------------------------------------------------
extern "C" void kernel_launch(void* const* d_in, const int* in_sizes, int n_in,
                              void* d_out, int out_size, void* d_ws, size_t ws_size,
                              hipStream_t stream) {
  const float* x       = (const float*)d_in[0];
  const int*   ei      = (const int*)d_in[1];
  const int*   et      = (const int*)d_in[2];
  const float* w_sl    = (const float*)d_in[3];
  const float* b_sl    = (const float*)d_in[4];
  const float* w1_n    = (const float*)d_in[5];
  const float* b1_n    = (const float*)d_in[6];
  const float* gamma_n = (const float*)d_in[7];
  const float* beta_n  = (const float*)d_in[8];
  const float* w2_n    = (const float*)d_in[9];
  const float* b2_n    = (const float*)d_in[10];
  const float* w1_d    = (const float*)d_in[11];
  const float* b1_d    = (const float*)d_in[12];
  const float* gamma_d = (const float*)d_in[13];
  const float* beta_d  = (const float*)d_in[14];
  const float* w2_d    = (const float*)d_in[15];
  const float* b2_d    = (const float*)d_in[16];
  const float* w_gat   = (const float*)d_in[17];
  const float* b_gat   = (const float*)d_in[18];

  const int M = in_sizes[0] / F;          // N nodes
  const int E = in_sizes[2];              // edges
  const long long NF = (long long)M * F;

  float* buf0  = (float*)d_ws;            // agg_n, later x_new_d
  float* buf1  = buf0 + NF;               // agg_d, later gate logits
  float* buf2  = buf1 + NF;               // h1_n -> act_n
  float* buf3  = buf2 + NF;               // h1_d -> act_d
  float* buf4  = buf3 + NF;               // x_new_x
  float* buf5  = buf4 + NF;               // x_new_n
  float* stats = buf5 + NF;               // [0:256)=branch n, [256:512)=branch d

  const int threads = 256;
  const int blocksNF = (int)((NF + threads - 1) / threads);

  init_kernel<<<blocksNF, threads, 0, stream>>>(x, buf0, buf1, stats, NF);

  long long sitems = (long long)E * 32;
  int sblocks = (int)((sitems + threads - 1) / threads);
  scatter_kernel<<<sblocks, threads, 0, stream>>>(x, ei, et, buf0, buf1, E);

  const int mtiles = (M + 15) / 16;       // 6250
  gemm128_kernel<<<mtiles, 256, 0, stream>>>(x,    w_sl, b_sl, buf4, M);
  gemm128_kernel<<<mtiles, 256, 0, stream>>>(buf0, w1_n, b1_n, buf2, M);
  gemm128_kernel<<<mtiles, 256, 0, stream>>>(buf1, w1_d, b1_d, buf3, M);

  const int stat_blocks = (M + 63) / 64;
  col_stats_kernel<<<stat_blocks, F, 0, stream>>>(buf2, stats, M);
  col_stats_kernel<<<stat_blocks, F, 0, stream>>>(buf3, stats + 256, M);

  const float invM = 1.0f / (float)M;
  bn_relu_kernel<<<blocksNF, threads, 0, stream>>>(buf2, stats,       gamma_n, beta_n, NF, invM);
  bn_relu_kernel<<<blocksNF, threads, 0, stream>>>(buf3, stats + 256, gamma_d, beta_d, NF, invM);

  gemm128_kernel<<<mtiles, 256, 0, stream>>>(buf2, w2_n, b2_n, buf5, M);
  gemm128_kernel<<<mtiles, 256, 0, stream>>>(buf3, w2_d, b2_d, buf0, M);

  gemm_gate_kernel<<<mtiles, 256, 0, stream>>>(buf4, buf5, buf0, w_gat, b_gat, buf1, M);

  finalize_kernel<<<M, F, 0, stream>>>(buf1, buf4, buf5, buf0, (float*)d_out);
}